// HungarianMatcher_66365834658116
// MI455X (gfx1250) — compile-verified
//
#include <hip/hip_runtime.h>
#include <hip/hip_bf16.h>

typedef __attribute__((ext_vector_type(2))) float v2f;
typedef __attribute__((ext_vector_type(8))) float v8f;

#define NC      91      // num classes
#define NT      1600    // num targets
#define NQTOT   14400   // 16 * 900 flattened queries
#define KSTEPS  23      // ceil(92 / 4) K-steps of V_WMMA_F32_16X16X4_F32
#define TT_ALL  100     // 1600 / 16 target tiles
#define TT_Y    4       // grid.y split
#define TT_PER  (TT_ALL / TT_Y)

#define W_BBOX  5.0f
#define W_GIOU  2.0f

__global__ __launch_bounds__(128)
void hungarian_cost_kernel(const float* __restrict__ logits,  // [14400, 91]
                           const float* __restrict__ pboxes,  // [14400, 4] cxcywh
                           const float* __restrict__ tboxes,  // [1600, 4]  cxcywh
                           const int*   __restrict__ tids,    // [1600]
                           float*       __restrict__ out)     // [14400, 1600]
{
    const int lane = threadIdx.x & 31;
    const int wave = threadIdx.x >> 5;
    const int hi   = lane >> 4;      // K/M half selector (wave32 WMMA layout)
    const int m    = lane & 15;      // A-matrix row owned by this lane
    const int n    = lane & 15;      // B/C/D column owned by this lane

    const long qtile = (long)blockIdx.x * 4 + wave;   // 0..899
    const long qbase = qtile * 16;

    __builtin_prefetch(tboxes, 0, 0);   // global_prefetch_b8: pull target table toward L2/L0

    // ---- softmax statistics (max, 1/sum(exp)) for the 16 queries of this tile.
    // Each lane keeps the stats of its own A-matrix row m.
    float mx_m = 0.0f, rden_m = 1.0f;
    for (int q = 0; q < 16; ++q) {
        const float* row = logits + (qbase + q) * NC;
        float l0 = row[lane];                                   // lane      < 91 always
        float l1 = row[lane + 32];                              // lane+32   < 91 always (<=63)
        float l2 = (lane + 64 < NC) ? row[lane + 64] : -3.0e38f;
        float vmax = fmaxf(fmaxf(l0, l1), l2);
        #pragma unroll
        for (int off = 16; off > 0; off >>= 1)
            vmax = fmaxf(vmax, __shfl_xor(vmax, off, 32));
        float s = __expf(l0 - vmax) + __expf(l1 - vmax) + __expf(l2 - vmax);
        #pragma unroll
        for (int off = 16; off > 0; off >>= 1)
            s += __shfl_xor(s, off, 32);
        if (m == q) { mx_m = vmax; rden_m = 1.0f / s; }
    }

    // ---- A fragments: softmax probs of this tile, WMMA f32 16x16x4 layout.
    // a[ki] holds row m, K = 4*ki + 2*hi + {0,1}.  K padded 91 -> 92 with zeros.
    v2f a[KSTEPS];
    const float* lrow = logits + (qbase + m) * NC;
    #pragma unroll
    for (int ki = 0; ki < KSTEPS; ++ki) {
        int k0 = 4 * ki + 2 * hi;
        a[ki].x = (k0     < NC) ? __expf(lrow[k0]     - mx_m) * rden_m : 0.0f;
        a[ki].y = (k0 + 1 < NC) ? __expf(lrow[k0 + 1] - mx_m) * rden_m : 0.0f;
    }

    // ---- predicted boxes for this lane's 8 output rows (M = v + 8*hi)
    float4 pb[8];
    #pragma unroll
    for (int v = 0; v < 8; ++v)
        pb[v] = ((const float4*)pboxes)[qbase + v + 8 * hi];

    const int tt0 = blockIdx.y * TT_PER;
    for (int tt = tt0; tt < tt0 + TT_PER; ++tt) {
        const int tbase = tt * 16;
        const float4 tb = ((const float4*)tboxes)[tbase + n];
        const int id_n  = tids[tbase + n];

        // cost_class tile via one-hot matmul: C[m][n] = P[m][id_n]   (exact, f32 WMMA)
        v8f c = {};
        #pragma unroll
        for (int ki = 0; ki < KSTEPS; ++ki) {
            const int koff = 4 * ki + 2 * hi;
            v2f b;
            b.x = (id_n == koff)     ? 1.0f : 0.0f;
            b.y = (id_n == koff + 1) ? 1.0f : 0.0f;
            c = __builtin_amdgcn_wmma_f32_16x16x4_f32(
                    /*neg_a=*/false, a[ki], /*neg_b=*/false, b,
                    /*c_mod=*/(short)0, c, /*reuse_a=*/false, /*reuse_b=*/false);
        }

        // target-box derived values (fixed per lane for this tile)
        const float tx1 = tb.x - 0.5f * tb.z, ty1 = tb.y - 0.5f * tb.w;
        const float tx2 = tb.x + 0.5f * tb.z, ty2 = tb.y + 0.5f * tb.w;
        const float areaT = tb.z * tb.w;

        #pragma unroll
        for (int v = 0; v < 8; ++v) {
            const float4 p = pb[v];
            // L1 on cxcywh
            float l1c = fabsf(p.x - tb.x) + fabsf(p.y - tb.y)
                      + fabsf(p.z - tb.z) + fabsf(p.w - tb.w);
            // GIoU on xyxy
            float px1 = p.x - 0.5f * p.z, py1 = p.y - 0.5f * p.w;
            float px2 = p.x + 0.5f * p.z, py2 = p.y + 0.5f * p.w;
            float areaP = p.z * p.w;
            float iw = fmaxf(fminf(px2, tx2) - fmaxf(px1, tx1), 0.0f);
            float ih = fmaxf(fminf(py2, ty2) - fmaxf(py1, ty1), 0.0f);
            float inter = iw * ih;
            float uni   = areaP + areaT - inter;
            float iou   = inter / uni;
            float cw = fmaxf(fmaxf(px2, tx2) - fminf(px1, tx1), 0.0f);
            float ch = fmaxf(fmaxf(py2, ty2) - fminf(py1, ty1), 0.0f);
            float areaC = cw * ch;
            float giou  = iou - (areaC - uni) / areaC;

            // C = 5*L1 + 1*(-prob) + 2*(-giou)
            float cost = W_BBOX * l1c - c[v] - W_GIOU * giou;
            out[(qbase + v + 8 * hi) * (long)NT + tbase + n] = cost;
        }
    }
}

extern "C" void kernel_launch(void* const* d_in, const int* in_sizes, int n_in,
                              void* d_out, int out_size, void* d_ws, size_t ws_size,
                              hipStream_t stream) {
    const float* logits = (const float*)d_in[0];   // pred_logits [16,900,91]
    const float* pboxes = (const float*)d_in[1];   // pred_boxes  [16,900,4]
    const float* tboxes = (const float*)d_in[2];   // tgt_bbox    [1600,4]
    const int*   tids   = (const int*)  d_in[3];   // tgt_ids     [1600]
    float* out = (float*)d_out;                    // [16,900,1600]

    dim3 grid(225, TT_Y, 1);   // 225 blocks * 4 waves = 900 query tiles; y splits target tiles
    dim3 block(128, 1, 1);     // 4 wave32
    hungarian_cost_kernel<<<grid, block, 0, stream>>>(logits, pboxes, tboxes, tids, out);
}